// PairwiseGNN_76776835383991
// MI455X (gfx1250) — compile-verified
//
#include <hip/hip_runtime.h>
#include <hip/hip_bf16.h>

typedef __attribute__((ext_vector_type(2))) float v2f;
typedef __attribute__((ext_vector_type(8))) float v8f;

#define HID 64
#define IN_CH 128

// ---------------- degree / normalization ----------------

__global__ void k_deg_init(float* deg, int n) {
    int i = blockIdx.x * blockDim.x + threadIdx.x;
    if (i < n) deg[i] = 1.0f;  // self loop
}

__global__ void k_deg_acc(const int* dst, float* deg, int e) {
    int i = blockIdx.x * blockDim.x + threadIdx.x;
    if (i < e) atomicAdd(&deg[dst[i]], 1.0f);
}

__global__ void k_dinv(float* deg, int n) {
    int i = blockIdx.x * blockDim.x + threadIdx.x;
    if (i < n) deg[i] = rsqrtf(deg[i]);  // in place: deg -> dinv
}

// ---------------- WMMA f32 GEMM: C[M x Nout] = A[M x K] * B[K x Nout] (+bias) ----------------
// One wave32 computes one 16x16 tile. M, K, Nout all multiples of 16.
// A fragment (16x4 f32, 2 VGPRs): lanes 0-15 -> M=lane,K base 0; lanes 16-31 -> M=lane-16,K base 2.
// B fragment (4x16 f32, 2 VGPRs): lane gives N=lane&15, same K-base split.
// C/D (16x16 f32, 8 VGPRs): VGPR r -> row r (lanes 0-15) / row r+8 (lanes 16-31), col = lane&15.

__global__ void k_wmma_gemm(const float* __restrict__ A, const float* __restrict__ B,
                            float* __restrict__ C, int mtiles, int ntiles, int K,
                            int nout, const float* __restrict__ bias, int has_bias) {
    int wave = (blockIdx.x * blockDim.x + threadIdx.x) >> 5;
    if (wave >= mtiles * ntiles) return;  // whole-wave uniform exit
    int mtile = wave / ntiles;
    int ntile = wave % ntiles;

    int lane = threadIdx.x & 31;
    int m    = lane & 15;
    int hi   = lane >> 4;        // 0 or 1
    int kb   = hi << 1;          // K base within group of 4: 0 or 2
    int n    = lane & 15;

    const float* Arow = A + (size_t)(mtile * 16 + m) * K;
    const float* Bcol = B + (size_t)(ntile * 16 + n);

    v8f acc = {};
    for (int k0 = 0; k0 < K; k0 += 4) {
        v2f a, b;
        a.x = Arow[k0 + kb];
        a.y = Arow[k0 + kb + 1];
        b.x = Bcol[(size_t)(k0 + kb) * nout];
        b.y = Bcol[(size_t)(k0 + kb + 1) * nout];
        acc = __builtin_amdgcn_wmma_f32_16x16x4_f32(
            /*neg_a=*/false, a, /*neg_b=*/false, b,
            /*c_mod=*/(short)0, acc, /*reuse_a=*/false, /*reuse_b=*/false);
    }

    int col = ntile * 16 + (lane & 15);
    int row = mtile * 16 + (hi << 3);
    float bv = has_bias ? bias[col] : 0.0f;
#pragma unroll
    for (int r = 0; r < 8; ++r)
        C[(size_t)(row + r) * nout + col] = acc[r] + bv;
}

// ---------------- aggregation ----------------

// agg[i][f] = dinv[i]^2 * h[i][f]   (self-loop contribution, initializes accumulator)
__global__ void k_self_init(const float* __restrict__ h, const float* __restrict__ dinv,
                            float* __restrict__ agg, int n) {
    int idx = blockIdx.x * blockDim.x + threadIdx.x;
    if (idx < n * HID) {
        int i = idx >> 6;  // /HID
        float d = dinv[i];
        agg[idx] = d * d * h[idx];
    }
}

// one wave32 per edge; each lane handles 2 of 64 features
__global__ void k_edge_scatter(const int* __restrict__ src, const int* __restrict__ dst,
                               const float* __restrict__ dinv, const float* __restrict__ h,
                               float* __restrict__ agg, int e) {
    int wave = (blockIdx.x * blockDim.x + threadIdx.x) >> 5;
    int lane = threadIdx.x & 31;
    if (wave >= e) return;
    int s = src[wave];
    int d = dst[wave];
    float w = dinv[s] * dinv[d];
    int f = lane << 1;
    const float2 hv = *(const float2*)(h + (size_t)s * HID + f);
    float* out = agg + (size_t)d * HID + f;
    atomicAdd(out + 0, w * hv.x);
    atomicAdd(out + 1, w * hv.y);
}

__global__ void k_bias_relu(const float* __restrict__ agg, const float* __restrict__ bias,
                            float* __restrict__ out, int n) {
    int idx = blockIdx.x * blockDim.x + threadIdx.x;
    if (idx < n * HID) {
        float v = agg[idx] + bias[idx & (HID - 1)];
        out[idx] = v > 0.0f ? v : 0.0f;
    }
}

// ---------------- launch ----------------

extern "C" void kernel_launch(void* const* d_in, const int* in_sizes, int n_in,
                              void* d_out, int out_size, void* d_ws, size_t ws_size,
                              hipStream_t stream) {
    const float* x    = (const float*)d_in[0];
    const int*   eidx = (const int*)d_in[1];   // jax default x64-disabled -> int32, [2,E] flat
    const float* W1   = (const float*)d_in[2];
    const float* b1   = (const float*)d_in[3];
    const float* W2   = (const float*)d_in[4];
    const float* b2   = (const float*)d_in[5];
    const float* Wd   = (const float*)d_in[6];
    const float* bd   = (const float*)d_in[7];

    const int N = in_sizes[0] / IN_CH;   // 100000
    const int E = in_sizes[1] / 2;       // 1600000
    const int* src = eidx;
    const int* dst = eidx + E;

    // workspace layout
    float* dinv = (float*)d_ws;                             // N floats
    float* bufA = dinv + ((N + 255) & ~255);                // N*HID floats
    float* bufB = bufA + (size_t)N * HID;                   // N*HID floats

    float* h_out   = (float*)d_out;                // [N, HID]
    float* dec_out = (float*)d_out + (size_t)N * HID;  // [N, IN_CH]

    const int TPB = 256;
    const int WPB = TPB / 32;  // 8 waves per block (wave32)

    // 1) normalization factors
    k_deg_init<<<(N + TPB - 1) / TPB, TPB, 0, stream>>>(dinv, N);
    k_deg_acc<<<(E + TPB - 1) / TPB, TPB, 0, stream>>>(dst, dinv, E);
    k_dinv<<<(N + TPB - 1) / TPB, TPB, 0, stream>>>(dinv, N);

    const int mtiles = N / 16;  // 6250 exact

    // 2) layer 1: t1 = x @ W1 -> bufA
    {
        int tiles = mtiles * (HID / 16);
        k_wmma_gemm<<<(tiles + WPB - 1) / WPB, TPB, 0, stream>>>(
            x, W1, bufA, mtiles, HID / 16, IN_CH, HID, nullptr, 0);
    }
    //    aggregate -> bufB, relu(+b1) -> bufA
    k_self_init<<<((size_t)N * HID + TPB - 1) / TPB, TPB, 0, stream>>>(bufA, dinv, bufB, N);
    k_edge_scatter<<<(E + WPB - 1) / WPB, TPB, 0, stream>>>(src, dst, dinv, bufA, bufB, E);
    k_bias_relu<<<((size_t)N * HID + TPB - 1) / TPB, TPB, 0, stream>>>(bufB, b1, bufA, N);

    // 3) layer 2: t2 = h1 @ W2 -> bufB
    {
        int tiles = mtiles * (HID / 16);
        k_wmma_gemm<<<(tiles + WPB - 1) / WPB, TPB, 0, stream>>>(
            bufA, W2, bufB, mtiles, HID / 16, HID, HID, nullptr, 0);
    }
    //    aggregate -> bufA, relu(+b2) -> d_out (h)
    k_self_init<<<((size_t)N * HID + TPB - 1) / TPB, TPB, 0, stream>>>(bufB, dinv, bufA, N);
    k_edge_scatter<<<(E + WPB - 1) / WPB, TPB, 0, stream>>>(src, dst, dinv, bufB, bufA, E);
    k_bias_relu<<<((size_t)N * HID + TPB - 1) / TPB, TPB, 0, stream>>>(bufA, b2, h_out, N);

    // 4) decoder: dec = h @ Wd + bd -> d_out[N*HID ...]
    {
        int tiles = mtiles * (IN_CH / 16);
        k_wmma_gemm<<<(tiles + WPB - 1) / WPB, TPB, 0, stream>>>(
            h_out, Wd, dec_out, mtiles, IN_CH / 16, HID, IN_CH, bd, 1);
    }
}